// MultiBoxLoss_14121852469982
// MI455X (gfx1250) — compile-verified
//
#include <hip/hip_runtime.h>
#include <hip/hip_bf16.h>

#define BB   128
#define PP   8732
#define CC   21
#define NGT  64
#define NBLK2 ((BB * PP) / 256)   // 4366, exact

typedef __attribute__((ext_vector_type(2))) float v2f;
typedef __attribute__((ext_vector_type(8))) float v8f;

// ---- wave32 sum via V_WMMA_F32_16X16X4_F32 -------------------------------
// A = 16x4 fp32 (2 VGPRs/lane), B = ones(4x16) => D[m,n] = v[lane m]+v[lane m+16].
// Lane-sum of the 8 accumulator rows + shfl_xor(16) yields the wave total.
__device__ __forceinline__ float wave_sum_wmma(float v) {
    v2f a; a.x = v;    a.y = 0.0f;
    v2f b; b.x = 1.0f; b.y = 1.0f;
    v8f c = {};
    c = __builtin_amdgcn_wmma_f32_16x16x4_f32(false, a, false, b,
                                              (short)0, c, false, false);
    float s = c[0] + c[1] + c[2] + c[3] + c[4] + c[5] + c[6] + c[7];
    s += __shfl_xor(s, 16, 32);
    return s;  // full-wave total in every lane
}

// Block-wide sum (result valid in all threads).
__device__ __forceinline__ float block_sum(float v, float* s8) {
    __syncthreads();
    float w = wave_sum_wmma(v);
    if ((threadIdx.x & 31) == 0) s8[threadIdx.x >> 5] = w;
    __syncthreads();
    float t = 0.0f;
    int nw = blockDim.x >> 5;
    for (int i = 0; i < nw; ++i) t += s8[i];
    return t;
}

// ---- CDNA5 async global->LDS helpers -------------------------------------
// LDS byte offset of a __shared__ object: generic LDS pointers carry the LDS
// offset in addr[31:0] (flat-LDS aperture), so truncation gives the DS address.
__device__ __forceinline__ unsigned lds_off(const void* p) {
    return (unsigned)(unsigned long long)p;
}
__device__ __forceinline__ void async_load_b32(unsigned lds_addr, const void* gaddr) {
    asm volatile("global_load_async_to_lds_b32 %0, %1, off"
                 :: "v"(lds_addr), "v"(gaddr) : "memory");
}
__device__ __forceinline__ void async_wait0() {
    asm volatile("s_wait_asynccnt 0" ::: "memory");
}

// ---- Kernel 1: matching + localization partials (one block per image) ----
__global__ __launch_bounds__(256) void match_kernel(
    const float* __restrict__ pred_locs,    // B,P,4
    const float* __restrict__ targets,      // B,N,6
    const float* __restrict__ prior_boxes,  // P,4 (cx,cy,w,h)
    int*   __restrict__ conf_t,             // B,P
    int*   __restrict__ best_idx,           // B,P
    float* __restrict__ best_ov,            // B,P
    float* __restrict__ loc_partial,        // B
    int*   __restrict__ npos_row)           // B
{
    __shared__ float sb[NGT][4];
    __shared__ float sarea[NGT];
    __shared__ int   slab[NGT];
    __shared__ int   sbest_prior[NGT];
    __shared__ float red_v[256];
    __shared__ int   red_i[256];
    __shared__ float s8[8];

    const int b = blockIdx.x, tid = threadIdx.x;
    const float* tg = targets + (size_t)b * NGT * 6;
    for (int g = tid; g < NGT; g += 256) {
        float x0 = tg[g*6+0], y0 = tg[g*6+1], x1 = tg[g*6+2], y1 = tg[g*6+3];
        sb[g][0] = x0; sb[g][1] = y0; sb[g][2] = x1; sb[g][3] = y1;
        sarea[g] = (x1 - x0) * (y1 - y0);
        slab[g]  = (int)tg[g*6+4];
    }
    __syncthreads();

    // Pass A: per prior, best GT (first-index argmax like jnp.argmax)
    for (int p = tid; p < PP; p += 256) {
        float pcx = prior_boxes[p*4+0], pcy = prior_boxes[p*4+1];
        float pw  = prior_boxes[p*4+2], ph  = prior_boxes[p*4+3];
        float bx0 = pcx - pw*0.5f, by0 = pcy - ph*0.5f;
        float bx1 = pcx + pw*0.5f, by1 = pcy + ph*0.5f;
        float parea = (bx1 - bx0) * (by1 - by0);
        float bestv = -1.0f; int bestg = 0;
        for (int g = 0; g < NGT; ++g) {
            float lx = fmaxf(sb[g][0], bx0), ly = fmaxf(sb[g][1], by0);
            float rx = fminf(sb[g][2], bx1), ry = fminf(sb[g][3], by1);
            float iw = fmaxf(rx - lx, 0.0f), ih = fmaxf(ry - ly, 0.0f);
            float inter = iw * ih;
            float iou = inter / (sarea[g] + parea - inter);
            if (iou > bestv) { bestv = iou; bestg = g; }
        }
        best_ov [(size_t)b*PP + p] = bestv;
        best_idx[(size_t)b*PP + p] = bestg;
    }

    // Pass B: per GT, best prior (4 threads/GT over contiguous quarters)
    {
        const int CH = PP / 4;           // 2183
        int g = tid >> 2, q = tid & 3;
        float gx0 = sb[g][0], gy0 = sb[g][1], gx1 = sb[g][2], gy1 = sb[g][3];
        float garea = sarea[g];
        float bestv = -1.0f; int bestp = q * CH;
        for (int p = q*CH; p < (q+1)*CH; ++p) {
            float pcx = prior_boxes[p*4+0], pcy = prior_boxes[p*4+1];
            float pw  = prior_boxes[p*4+2], ph  = prior_boxes[p*4+3];
            float bx0 = pcx - pw*0.5f, by0 = pcy - ph*0.5f;
            float bx1 = pcx + pw*0.5f, by1 = pcy + ph*0.5f;
            float lx = fmaxf(gx0, bx0), ly = fmaxf(gy0, by0);
            float rx = fminf(gx1, bx1), ry = fminf(gy1, by1);
            float iw = fmaxf(rx - lx, 0.0f), ih = fmaxf(ry - ly, 0.0f);
            float inter = iw * ih;
            float iou = inter / (garea + (bx1-bx0)*(by1-by0) - inter);
            if (iou > bestv) { bestv = iou; bestp = p; }
        }
        red_v[tid] = bestv; red_i[tid] = bestp;
    }
    __syncthreads();
    if (tid < NGT) {
        float bv = red_v[tid*4]; int bp = red_i[tid*4];
        for (int q = 1; q < 4; ++q)
            if (red_v[tid*4+q] > bv) { bv = red_v[tid*4+q]; bp = red_i[tid*4+q]; }
        sbest_prior[tid] = bp;
    }
    __syncthreads();
    if (tid == 0) {  // sequential scatter: last-write-wins like jnp .at[].set
        for (int n = 0; n < NGT; ++n) {
            int p = sbest_prior[n];
            best_idx[(size_t)b*PP + p] = n;
            best_ov [(size_t)b*PP + p] = 1.0f;
        }
    }
    __threadfence_block();
    __syncthreads();

    // Pass C: labels, encode, L1 partials
    float loc_sum = 0.0f; float npos = 0.0f;
    for (int p = tid; p < PP; p += 256) {
        float bo = best_ov [(size_t)b*PP + p];
        int   bi = best_idx[(size_t)b*PP + p];
        int lbl = (bo < 0.5f) ? 0 : slab[bi];
        conf_t[(size_t)b*PP + p] = lbl;
        if (lbl > 0) {
            npos += 1.0f;
            float x0 = sb[bi][0], y0 = sb[bi][1], x1 = sb[bi][2], y1 = sb[bi][3];
            float cx = (x0 + x1) * 0.5f, cy = (y0 + y1) * 0.5f;
            float w  = x1 - x0,          h  = y1 - y0;
            float pcx = prior_boxes[p*4+0], pcy = prior_boxes[p*4+1];
            float pw  = prior_boxes[p*4+2], ph  = prior_boxes[p*4+3];
            float e0 = (cx - pcx) / (pw * 0.1f);
            float e1 = (cy - pcy) / (ph * 0.1f);
            float e2 = logf(w / pw + 1e-10f) * 5.0f;
            float e3 = logf(h / ph + 1e-10f) * 5.0f;
            const float* pl = pred_locs + ((size_t)b*PP + p) * 4;
            loc_sum += fabsf(pl[0]-e0) + fabsf(pl[1]-e1)
                     + fabsf(pl[2]-e2) + fabsf(pl[3]-e3);
        }
    }
    float ls = block_sum(loc_sum, s8);
    float ns = block_sum(npos,   s8);
    if (tid == 0) { loc_partial[b] = ls; npos_row[b] = (int)(ns + 0.5f); }
}

// ---- Kernel 2: log-softmax CE; async-staged rows; pos partials + neg buf --
__global__ __launch_bounds__(256) void ce_kernel(
    const float* __restrict__ pred_confs,   // B,P,C
    const int*   __restrict__ conf_t,       // B,P
    float* __restrict__ negbuf,             // B,P
    float* __restrict__ pos_partial)        // NBLK2
{
    __shared__ float xs[256 * CC];          // 21,504 B, block's 256 rows
    __shared__ float s8[8];
    const int tid = threadIdx.x;
    size_t base = (size_t)blockIdx.x * 256;
    const float* src = pred_confs + base * CC;

    // Coalesced, dword-granular async DMA into LDS (ASYNCcnt-tracked).
    #pragma unroll
    for (int i = 0; i < CC; ++i) {
        int off = tid + i * 256;            // contiguous across the block
        async_load_b32(lds_off(&xs[off]), src + off);
    }
    async_wait0();
    __syncthreads();

    const float* x = &xs[tid * CC];
    float m = x[0];
    #pragma unroll
    for (int c = 1; c < CC; ++c) m = fmaxf(m, x[c]);
    float se = 0.0f;
    #pragma unroll
    for (int c = 0; c < CC; ++c) se += expf(x[c] - m);
    float lse = m + logf(se);
    size_t idx = base + tid;
    int lbl = conf_t[idx];
    float ce = lse - x[lbl];
    float posv, negv;
    if (lbl > 0) { posv = ce;   negv = 0.0f; }
    else         { posv = 0.0f; negv = fmaxf(ce, 0.0f); }  // ce >= 0 up to rounding
    negbuf[idx] = negv;
    float ps = block_sum(posv, s8);
    if (tid == 0) pos_partial[blockIdx.x] = ps;
}

// ---- Kernel 3: per-row top-(3*n_pos) sum via LDS-resident radix-select ----
__global__ __launch_bounds__(256) void topk_kernel(
    const float* __restrict__ negbuf,     // B,P (all >= 0)
    const int*   __restrict__ npos_row,   // B
    float* __restrict__ topk_partial)     // B
{
    __shared__ float    xv[PP];           // 34,928 B row cache
    __shared__ unsigned hist[256];
    __shared__ float    s8[8];
    const int b = blockIdx.x, tid = threadIdx.x;
    const float* v = negbuf + (size_t)b * PP;

    // Stage the whole row into LDS once (5 passes then read LDS, not HBM).
    for (int p = tid; p < PP; p += 256)
        async_load_b32(lds_off(&xv[p]), v + p);
    async_wait0();
    __syncthreads();

    int k = npos_row[b] * 3;
    if (k > PP) k = PP;
    if (k <= 0) { if (tid == 0) topk_partial[b] = 0.0f; return; }

    unsigned prefix = 0;
    int remaining = k;
    for (int pass = 0; pass < 4; ++pass) {
        int shift = 24 - pass * 8;
        unsigned himask = (pass == 0) ? 0u : (0xFFFFFFFFu << (shift + 8));
        hist[tid] = 0;
        __syncthreads();
        for (int p = tid; p < PP; p += 256) {
            unsigned u = __float_as_uint(xv[p]);
            if ((u & himask) == prefix) atomicAdd(&hist[(u >> shift) & 255u], 1u);
        }
        __syncthreads();
        unsigned cum = 0; int sel = 0;          // redundant walk, all threads
        for (int bin = 255; bin >= 0; --bin) {
            unsigned h = hist[bin];
            if (cum + h >= (unsigned)remaining) { sel = bin; break; }
            cum += h;
        }
        remaining -= (int)cum;
        prefix |= ((unsigned)sel) << shift;
        __syncthreads();
    }
    // k-th largest has bit pattern == prefix; exact top-k sum with tie fixup
    float tF = __uint_as_float(prefix);
    float sgt = 0.0f, cgt = 0.0f;
    for (int p = tid; p < PP; p += 256) {
        float f = xv[p];
        if (__float_as_uint(f) > prefix) { sgt += f; cgt += 1.0f; }
    }
    float S = block_sum(sgt, s8);
    float Cn = block_sum(cgt, s8);
    if (tid == 0) topk_partial[b] = S + ((float)k - Cn) * tF;
}

// ---- Kernel 4: finalize ---------------------------------------------------
__global__ __launch_bounds__(256) void finalize_kernel(
    const float* __restrict__ loc_partial,   // B
    const int*   __restrict__ npos_row,      // B
    const float* __restrict__ pos_partial,   // NBLK2
    const float* __restrict__ topk_partial,  // B
    float* __restrict__ out)                 // [conf_loss, loc_loss]
{
    __shared__ float s8[8];
    const int tid = threadIdx.x;
    float a = 0.0f, n = 0.0f, p = 0.0f, t = 0.0f;
    for (int i = tid; i < BB; i += 256) {
        a += loc_partial[i];
        n += (float)npos_row[i];
        t += topk_partial[i];
    }
    for (int i = tid; i < NBLK2; i += 256) p += pos_partial[i];
    float loc_tot  = block_sum(a, s8);
    float npos_tot = block_sum(n, s8);
    float pos_tot  = block_sum(p, s8);
    float neg_tot  = block_sum(t, s8);
    if (tid == 0) {
        out[0] = (neg_tot + pos_tot) / npos_tot;          // conf_loss
        out[1] = loc_tot / (npos_tot * 4.0f);             // ALPHA * loc_loss
    }
}

extern "C" void kernel_launch(void* const* d_in, const int* in_sizes, int n_in,
                              void* d_out, int out_size, void* d_ws, size_t ws_size,
                              hipStream_t stream) {
    const float* pred_locs   = (const float*)d_in[0];
    const float* pred_confs  = (const float*)d_in[1];
    const float* targets     = (const float*)d_in[2];
    const float* prior_boxes = (const float*)d_in[3];
    float* out = (float*)d_out;

    char* w = (char*)d_ws;
    int*   conf_t = (int*)w;   w += (size_t)BB * PP * sizeof(int);
    int*   bidx   = (int*)w;   w += (size_t)BB * PP * sizeof(int);
    float* bov    = (float*)w; w += (size_t)BB * PP * sizeof(float); // reused as negbuf
    float* locp   = (float*)w; w += (size_t)BB * sizeof(float);
    int*   nposr  = (int*)w;   w += (size_t)BB * sizeof(int);
    float* posp   = (float*)w; w += (size_t)NBLK2 * sizeof(float);
    float* tkp    = (float*)w;

    match_kernel   <<<BB,    256, 0, stream>>>(pred_locs, targets, prior_boxes,
                                               conf_t, bidx, bov, locp, nposr);
    ce_kernel      <<<NBLK2, 256, 0, stream>>>(pred_confs, conf_t, bov, posp);
    topk_kernel    <<<BB,    256, 0, stream>>>(bov, nposr, tkp);
    finalize_kernel<<<1,     256, 0, stream>>>(locp, nposr, posp, tkp, out);
}